// GaugeField_63058709840600
// MI455X (gfx1250) — compile-verified
//
#include <hip/hip_runtime.h>
#include <stdint.h>

#define EPS_EXP 1e-8f
#define EPS_LOG 1e-4f

// ---- CDNA5 async LDS DMA (ASYNCcnt) helpers -------------------------------
// GVS mode: 64-bit scalar base in SGPR pair + 32-bit byte offset in a VGPR.
__device__ __forceinline__ uint32_t lds_off(const void* p) {
  // Generic LDS pointer: addr[31:0] is the group-segment byte offset (ISA 10.2).
  return (uint32_t)(uintptr_t)p;
}
__device__ __forceinline__ void async_ld_b32(uint32_t lds, const void* base, uint32_t voff) {
  asm volatile("global_load_async_to_lds_b32 %0, %1, %2"
               :: "v"(lds), "v"(voff), "s"(base) : "memory");
}
__device__ __forceinline__ void async_ld_b128(uint32_t lds, const void* base, uint32_t voff) {
  asm volatile("global_load_async_to_lds_b128 %0, %1, %2"
               :: "v"(lds), "v"(voff), "s"(base) : "memory");
}
__device__ __forceinline__ void async_st_b32(const void* base, uint32_t voff, uint32_t lds) {
  asm volatile("global_store_async_from_lds_b32 %0, %1, %2"
               :: "v"(voff), "v"(lds), "s"(base) : "memory");
}
__device__ __forceinline__ void async_st_b128(const void* base, uint32_t voff, uint32_t lds) {
  asm volatile("global_store_async_from_lds_b128 %0, %1, %2"
               :: "v"(voff), "v"(lds), "s"(base) : "memory");
}
__device__ __forceinline__ void wait_async0() {
  asm volatile("s_wait_asynccnt 0x0" ::: "memory");
}

// ---- 3x3 math -------------------------------------------------------------
// Rodrigues exp of the skew part of omega[e]; sign<0 applies transpose
// (exp(S)^T == exp(-S) -> negate axis). Only the 6 off-diagonals are read.
__device__ __forceinline__ void rod_exp(const float* __restrict__ om,
                                        uint32_t e, int s, float U[9]) {
  const float* o = om + e * 9u;
  float o1 = o[1], o2 = o[2], o3 = o[3], o5 = o[5], o6 = o[6], o7 = o[7];
  float wx = 0.5f * (o7 - o5);   // S[2][1]
  float wy = 0.5f * (o2 - o6);   // S[0][2]
  float wz = 0.5f * (o3 - o1);   // S[1][0]
  if (s < 0) { wx = -wx; wy = -wy; wz = -wz; }
  float t2 = wx * wx + wy * wy + wz * wz;
  float th = sqrtf(t2);
  float a, b;
  if (th < EPS_EXP) {
    a = 1.0f - t2 * (1.0f / 6.0f);
    b = 0.5f - t2 * (1.0f / 24.0f);
  } else {
    float sn = sinf(th), cs = cosf(th);
    a = sn / th;
    b = (1.0f - cs) / t2;
  }
  // U = I + a*S + b*S^2, with S^2 = w w^T - |w|^2 I (exact for skew S)
  U[0] = 1.0f + b * (wx * wx - t2);
  U[1] = b * wx * wy - a * wz;
  U[2] = b * wx * wz + a * wy;
  U[3] = b * wy * wx + a * wz;
  U[4] = 1.0f + b * (wy * wy - t2);
  U[5] = b * wy * wz - a * wx;
  U[6] = b * wz * wx - a * wy;
  U[7] = b * wz * wy + a * wx;
  U[8] = 1.0f + b * (wz * wz - t2);
}

__device__ __forceinline__ void mm3(const float A[9], const float B[9], float C[9]) {
#pragma unroll
  for (int i = 0; i < 3; ++i)
#pragma unroll
    for (int j = 0; j < 3; ++j)
      C[i * 3 + j] = A[i * 3 + 0] * B[0 * 3 + j]
                   + A[i * 3 + 1] * B[1 * 3 + j]
                   + A[i * 3 + 2] * B[2 * 3 + j];
}

// ---- fused kernel: gather -> exp -> triple product -> log -----------------
__global__ __launch_bounds__(256) void gauge_fused(
    const float* __restrict__ omega, const int* __restrict__ idx,
    const int* __restrict__ sgn, float* __restrict__ out, int T) {
  __shared__ alignas(16) int   s_idx[256 * 3];
  __shared__ alignas(16) int   s_sgn[256 * 3];
  __shared__ alignas(16) float s_out[256 * 9];

  const uint32_t tid  = threadIdx.x;
  const uint32_t tot3 = 3u * (uint32_t)T;
  const uint32_t blk3 = blockIdx.x * 768u;

  // Coalesced async b128 DMA of this block's idx/sign slab into LDS
  // (192 vec4s per array; b32 fallback for a ragged global tail).
  if (tid < 192u) {
    uint32_t g = blk3 + tid * 4u;                 // dword index
    if (g + 4u <= tot3) {
      async_ld_b128(lds_off(&s_idx[tid * 4u]), idx, g * 4u);
      async_ld_b128(lds_off(&s_sgn[tid * 4u]), sgn, g * 4u);
    } else {
#pragma unroll
      for (uint32_t k = 0; k < 4u; ++k)
        if (g + k < tot3) {
          async_ld_b32(lds_off(&s_idx[tid * 4u + k]), idx, (g + k) * 4u);
          async_ld_b32(lds_off(&s_sgn[tid * 4u + k]), sgn, (g + k) * 4u);
        }
    }
  }
  wait_async0();
  __syncthreads();

  const uint32_t t = blockIdx.x * 256u + tid;
  if (t < (uint32_t)T) {
    float U1[9], U2[9], M[9], U3[9], H[9];
    rod_exp(omega, (uint32_t)s_idx[tid * 3u + 0], s_sgn[tid * 3u + 0], U1);
    rod_exp(omega, (uint32_t)s_idx[tid * 3u + 1], s_sgn[tid * 3u + 1], U2);
    mm3(U2, U1, M);                                   // M = U2*U1
    rod_exp(omega, (uint32_t)s_idx[tid * 3u + 2], s_sgn[tid * 3u + 2], U3);
    mm3(U3, M, H);                                    // H = U3*U2*U1

    // log_so3
    float tr = H[0] + H[4] + H[8];
    float ct = (tr - 1.0f) * 0.5f;
    ct = fminf(fmaxf(ct, -1.0f + 1e-6f), 1.0f - 1e-6f);
    float th = acosf(ct);
    float f;
    if (th < EPS_LOG) f = 0.5f + th * th * (1.0f / 12.0f);
    else              f = th / (2.0f * sinf(th));
    float a01 = f * (H[1] - H[3]);
    float a02 = f * (H[2] - H[6]);
    float a12 = f * (H[5] - H[7]);

    float* so = &s_out[tid * 9u];
    so[0] = 0.0f; so[1] = a01;  so[2] = a02;
    so[3] = -a01; so[4] = 0.0f; so[5] = a12;
    so[6] = -a02; so[7] = -a12; so[8] = 0.0f;
  }
  __syncthreads();

  // Coalesced async b128 DMA of the 9-float results back out
  // (576 vec4s = 2304 dwords per block; b32 fallback on the ragged tail).
  const uint32_t blk9 = blockIdx.x * 2304u;
  const uint32_t tot9 = 9u * (uint32_t)T;
#pragma unroll
  for (uint32_t j = 0; j < 3u; ++j) {
    uint32_t v = j * 256u + tid;                  // vec4 index, 0..575
    if (v < 576u) {
      uint32_t g = blk9 + v * 4u;                 // dword index
      if (g + 4u <= tot9) {
        async_st_b128(out, g * 4u, lds_off(&s_out[v * 4u]));
      } else {
#pragma unroll
        for (uint32_t k = 0; k < 4u; ++k)
          if (g + k < tot9)
            async_st_b32(out, (g + k) * 4u, lds_off(&s_out[v * 4u + k]));
      }
    }
  }
  // S_ENDPGM performs an implicit wait-idle, draining ASYNCcnt.
}

extern "C" void kernel_launch(void* const* d_in, const int* in_sizes, int n_in,
                              void* d_out, int out_size, void* d_ws, size_t ws_size,
                              hipStream_t stream) {
  const float* omega = (const float*)d_in[0];
  const int*   idx   = (const int*)d_in[1];
  const int*   sgn   = (const int*)d_in[2];
  float*       out   = (float*)d_out;
  const int T = in_sizes[1] / 3;  // tri_edge_idx is (T,3)
  const int blocks = (T + 255) / 256;
  gauge_fused<<<blocks, 256, 0, stream>>>(omega, idx, sgn, out, T);
}